// ONNXVariableROIAlign_29781303231111
// MI455X (gfx1250) — compile-verified
//
#include <hip/hip_runtime.h>

#define RATIO 2

// One 16-byte LDS record per sample line (y or x), with the edge clamp and
// validity folded into the pair weights:
//   value = wa * f[idx] + wb * f[idx + 1]
struct alignas(16) Tap {
    int   idx;   // base index, clamped to [0, dim-2]
    float wa;    // weight for element idx
    float wb;    // weight for element idx+1
    int   pad;
};

// One block handles one ROI x one chunk of CH channels for a P x P output level.
// Threads: 256 (8 wave32). LDS holds 2*P*RATIO Tap records (y taps then x taps).
// DIM (= H = W), STRIDE and C are compile-time so all offset math reduces to
// shifts and 32-bit VGPR offsets off a uniform SGPR base.
template <int P, int CH, int DIM, int STRIDE>
__global__ __launch_bounds__(256) void roi_align_level_kernel(
    const float* __restrict__ feat,   // (B, 256, DIM, DIM)
    const float* __restrict__ rois,   // (N, 5) [b, x1, y1, x2, y2]
    float* __restrict__ out)          // (N, 256, P, P)
{
    constexpr int   S     = P * RATIO;        // sample lines per axis
    constexpr int   HW    = DIM * DIM;
    constexpr int   C     = 256;
    constexpr float SCALE = 1.0f / (float)STRIDE;

    __shared__ Tap   s_tap[2 * S];            // [0,S): y taps, [S,2S): x taps
    __shared__ float s_roi[5];

    const int n  = blockIdx.x;
    const int c0 = blockIdx.y * CH;
    const int t  = threadIdx.x;

    // Stage the ROI row once (prefetch -> LDS) instead of 2*2S scattered reads.
    if (t == 0) __builtin_prefetch(rois + n * 5, 0, 3);   // global_prefetch
    if (t < 5)  s_roi[t] = rois[n * 5 + t];
    __syncthreads();

    // ---- Build per-ROI interpolation tables in LDS (first 2S threads) ----
    if (t < 2 * S) {
        const bool isY = (t < S);
        const int  s   = isY ? t : (t - S);

        const float lo  = s_roi[isY ? 2 : 1] * SCALE - 0.5f;
        const float hi  = s_roi[isY ? 4 : 3] * SCALE - 0.5f;
        const float bin = (hi - lo) / (float)P;

        const int   o   = s / RATIO;
        const int   r   = s % RATIO;
        const float g   = (float)o + ((float)r + 0.5f) / (float)RATIO;
        const float pos = lo + g * bin;

        const bool  valid = (pos > -1.0f) && (pos < (float)DIM);
        const float pc = fminf(fmaxf(pos, 0.0f), (float)(DIM - 1));
        const int   p0 = (int)floorf(pc);
        const float l  = pc - (float)p0;
        const float h  = 1.0f - l;

        Tap tp;
        if (p0 < DIM - 1) { tp.idx = p0;       tp.wa = h;    tp.wb = l;     }
        else              { tp.idx = DIM - 2;  tp.wa = 0.0f; tp.wb = h + l; }
        if (!valid)       { tp.wa = 0.0f;      tp.wb = 0.0f; }
        tp.pad = 0;
        s_tap[t] = tp;
    }
    __syncthreads();

    const int b = (int)s_roi[0];
    const float* __restrict__ fb = feat + ((size_t)(b * C + c0) * HW);
    float* __restrict__ ob = out + ((size_t)(n * C + c0) * (P * P));

    constexpr int total = CH * P * P;
    for (int e = t; e < total; e += 256) {
        const int pw = e % P;                  // constant-divisor -> mul/shift
        const int ph = (e / P) % P;
        const int c  = e / (P * P);

        float acc = 0.0f;
        #pragma unroll
        for (int iy = 0; iy < RATIO; ++iy) {
            const Tap ty = s_tap[ph * RATIO + iy];          // ds_load_b128
            const unsigned rowOff = (unsigned)c * HW + (unsigned)ty.idx * DIM;
            #pragma unroll
            for (int ix = 0; ix < RATIO; ++ix) {
                const Tap tx = s_tap[S + pw * RATIO + ix];  // ds_load_b128
                float2 a, bb;                               // one b64 per row
                __builtin_memcpy(&a,  fb + rowOff + tx.idx,       8);
                __builtin_memcpy(&bb, fb + rowOff + DIM + tx.idx, 8);
                const float hx0 = tx.wa * a.x  + tx.wb * a.y;
                const float hx1 = tx.wa * bb.x + tx.wb * bb.y;
                acc += ty.wa * hx0 + ty.wb * hx1;
            }
        }
        // Output is write-once / never re-read: non-temporal store so the
        // 176 MB of feature maps stay resident in the 192 MB L2.
        __builtin_nontemporal_store(acc * 0.25f, ob + e);
    }
}

extern "C" void kernel_launch(void* const* d_in, const int* in_sizes, int n_in,
                              void* d_out, int out_size, void* d_ws, size_t ws_size,
                              hipStream_t stream) {
    const float* feat1 = (const float*)d_in[0];   // (2,256,256,256)
    const float* feat2 = (const float*)d_in[1];   // (2,256,128,128)
    const float* feat3 = (const float*)d_in[2];   // (2,256,64,64)
    const float* rois  = (const float*)d_in[3];   // (N,5)

    const int N = in_sizes[3] / 5;                // 128
    const int C = 256;

    float* out1 = (float*)d_out;
    float* out2 = out1 + (size_t)N * C * 28 * 28;
    float* out3 = out2 + (size_t)N * C * 14 * 14;

    dim3 blk(256);
    // feat1: stride 4, P=28. 128 x 8 = 1024 blocks.
    roi_align_level_kernel<28, 32, 256, 4><<<dim3(N, C / 32), blk, 0, stream>>>(
        feat1, rois, out1);
    // feat2: stride 8, P=14. 128 x 16 = 2048 blocks.
    roi_align_level_kernel<14, 16, 128, 8><<<dim3(N, C / 16), blk, 0, stream>>>(
        feat2, rois, out2);
    // feat3: stride 16, P=7. 128 x 32 = 4096 blocks.
    roi_align_level_kernel<7, 8, 64, 16><<<dim3(N, C / 8), blk, 0, stream>>>(
        feat3, rois, out3);
}